// IDLHead_14766097563720
// MI455X (gfx1250) — compile-verified
//
#include <hip/hip_runtime.h>
#include <hip/hip_bf16.h>

typedef __attribute__((ext_vector_type(2))) float v2f;
typedef __attribute__((ext_vector_type(8))) float v8f;
typedef __attribute__((ext_vector_type(4))) unsigned int u32x4;
typedef __attribute__((ext_vector_type(4))) int i32x4;
typedef __attribute__((ext_vector_type(8))) int i32x8;

#define HS    64
#define TLEN  1024
#define BATCH 16
#define NKQV  192           // only cols 0..191 of pre are ever used
#define MROWS (BATCH * TLEN)

#if __has_builtin(__builtin_amdgcn_s_wait_tensorcnt) && __has_builtin(__builtin_amdgcn_tensor_load_to_lds)
#define HAVE_TDM 1
#define tdm_wait(n) __builtin_amdgcn_s_wait_tensorcnt(n)
#else
#define HAVE_TDM 0
#define tdm_wait(n) ((void)0)
#endif

// ---------------------------------------------------------------------------
// TDM 2D tile load: global (tile_d1 rows x tile_d0 elems, row pitch stride_d0
// f32 elements) -> LDS at lds_dst, with TDM-applied row padding of
// (pad_amt_code+1) dwords every 2^(pad_int_code+1) dwords (== tile_d0 here),
// i.e. LDS row stride = tile_d0 + pad_amt_code + 1 floats.
// Issue is wave-uniform; EXEC ignored by hardware; tracked by TENSORcnt.
// ---------------------------------------------------------------------------
__device__ __forceinline__ void tdm_load_2d(float* lds_dst, const void* gsrc,
                                            unsigned tile_d0, unsigned tile_d1,
                                            unsigned stride_d0,
                                            unsigned pad_int_code, unsigned pad_amt_code,
                                            int lane) {
#if HAVE_TDM
  const unsigned lds_off = (unsigned)(unsigned long long)lds_dst;  // addr[31:0] == LDS offset
  const unsigned long long ga = (unsigned long long)gsrc;
  u32x4 g0;
  g0[0] = 1u;                                               // count=1, user mode, no gather
  g0[1] = lds_off;                                          // lds_addr
  g0[2] = (unsigned)ga;                                     // global_addr[31:0]
  g0[3] = (unsigned)((ga >> 32) & 0x01ffffffu) | 0x80000000u; // global_addr[56:32] | type=2
  i32x8 g1;
  g1[0] = (int)((2u << 16)                 // data_size = 4B
              | (1u << 20)                 // pad_enable
              | (pad_int_code << 22) | (pad_amt_code << 25));
  g1[1] = (int)((tile_d0 & 0xffffu) << 16);                 // tensor_dim0[15:0]
  g1[2] = (int)((tile_d0 >> 16) | ((tile_d1 & 0xffffu) << 16)); // dim0[31:16] | dim1[15:0]
  g1[3] = (int)((tile_d1 >> 16) | ((tile_d0 & 0xffffu) << 16)); // dim1[31:16] | tile_dim0
  g1[4] = (int)(tile_d1 & 0xffffu);                         // tile_dim1 (tile_dim2 = 0)
  g1[5] = (int)stride_d0;                                   // tensor_dim0_stride[31:0]
  g1[6] = 0;                                                // stride[47:32] | dim1_stride lo
  g1[7] = 0;
  i32x4 gz = {0, 0, 0, 0};
#if defined(__clang_major__) && (__clang_major__ >= 23)
  i32x8 gz8 = {0, 0, 0, 0, 0, 0, 0, 0};
  __builtin_amdgcn_tensor_load_to_lds(g0, g1, gz, gz, gz8, 0);
#else
  __builtin_amdgcn_tensor_load_to_lds(g0, g1, gz, gz, 0);
#endif
#else
  // wave-cooperative fallback (synchronous), same padded layout
  const float* src = (const float*)gsrc;
  const unsigned pitch = tile_d0 + pad_amt_code + 1;
  for (unsigned i = (unsigned)lane; i < tile_d0 * tile_d1; i += 32) {
    const unsigned r = i / tile_d0, c = i % tile_d0;
    lds_dst[r * pitch + c] = src[(size_t)r * stride_d0 + c];
  }
#endif
}

// ---------------------------------------------------------------------------
// Kernel 0: An = A * (0.95 / max_t sum_j |A[t][j]|)
// ---------------------------------------------------------------------------
__global__ __launch_bounds__(256) void normalize_A(const float* __restrict__ A,
                                                   float* __restrict__ An) {
  __shared__ float red[256];
  const int t = threadIdx.x;
  float s = 0.f;
  #pragma unroll 8
  for (int j = 0; j < 256; ++j) s += fabsf(A[t * 256 + j]);
  red[t] = s;
  __syncthreads();
  for (int off = 128; off > 0; off >>= 1) {
    if (t < off) red[t] = fmaxf(red[t], red[t + off]);
    __syncthreads();
  }
  const float scale = 0.95f / red[0];
  #pragma unroll 8
  for (int j = 0; j < 256; ++j) An[t * 256 + j] = A[t * 256 + j] * scale;
}

// ---------------------------------------------------------------------------
// FP32 WMMA GEMM: out[M x 192] = Xsplit[M x K] @ W[:, 0:192] (+ bias)
//   Xsplit cols [0,K0) from A0 (stride lda0), cols [K0,K) from A1 (lda1).
//   K0 and K are multiples of 32, so every 32-wide K-chunk has one source.
// Block tile 64x64, 8 waves (16x32 each). Tiles staged by TDM, double-buffered.
// ---------------------------------------------------------------------------
__global__ __launch_bounds__(256) void gemm_wmma_f32(
    const float* __restrict__ A0, int lda0, int K0,
    const float* __restrict__ A1, int lda1,
    const float* __restrict__ W, int ldw,
    const float* __restrict__ bias,
    float* __restrict__ out, int K) {
  __shared__ float As[2][64][36];   // TDM pad: 32 + 4  (conflict-free A frags)
  __shared__ float Bs[2][32][65];   // TDM pad: 64 + 1

  const int lane = threadIdx.x & 31;
  const int wave = threadIdx.x >> 5;
  const int half = lane >> 4;
  const int lr   = lane & 15;
  const int wm   = wave & 3;   // 4 waves along M
  const int wn   = wave >> 2;  // 2 waves along N
  const int mbase = blockIdx.y * 64;
  const int nbase = blockIdx.x * 64;
  const int nchunks = K >> 5;

  v8f c0 = {}; v8f c1 = {};

  auto issue = [&](int c) {
    const int kt = c << 5;
    const float* asrc;
    int lda;
    if (kt < K0) { asrc = A0 + (size_t)mbase * lda0 + kt;        lda = lda0; }
    else         { asrc = A1 + (size_t)mbase * lda1 + (kt - K0); lda = lda1; }
    // A tile: 64 rows x 32 contiguous k      -> As[buf], LDS stride 36
    tdm_load_2d(&As[c & 1][0][0], asrc, 32, 64, (unsigned)lda, 4, 3, lane);
    // B tile: 32 rows x 64 contiguous n      -> Bs[buf], LDS stride 65
    tdm_load_2d(&Bs[c & 1][0][0], W + (size_t)kt * ldw + nbase, 64, 32, (unsigned)ldw, 5, 0, lane);
  };

  if (wave == 0) issue(0);
  for (int c = 0; c < nchunks; ++c) {
    if (wave == 0) {
      if (c + 1 < nchunks) { issue(c + 1); tdm_wait(2); }  // cur pair retired (in-order)
      else                 { tdm_wait(0); }
    }
    __syncthreads();
    const int buf = c & 1;
    #pragma unroll
    for (int kk = 0; kk < 8; ++kk) {
      const int k0 = kk * 4 + 2 * half;  // lane-half picks K offset per ISA layout
      v2f a;  a.x  = As[buf][wm * 16 + lr][k0];      a.y  = As[buf][wm * 16 + lr][k0 + 1];
      v2f b0; b0.x = Bs[buf][k0][wn * 32 + lr];      b0.y = Bs[buf][k0 + 1][wn * 32 + lr];
      v2f b1; b1.x = Bs[buf][k0][wn * 32 + 16 + lr]; b1.y = Bs[buf][k0 + 1][wn * 32 + 16 + lr];
      c0 = __builtin_amdgcn_wmma_f32_16x16x4_f32(false, a, false, b0, (short)0, c0, false, false);
      c1 = __builtin_amdgcn_wmma_f32_16x16x4_f32(false, a, false, b1, (short)0, c1, false, false);
    }
    __syncthreads();
  }

  #pragma unroll
  for (int v = 0; v < 8; ++v) {
    const int m  = mbase + wm * 16 + v + 8 * half;
    const int n0 = nbase + wn * 32 + lr;
    float x0 = c0[v], x1 = c1[v];
    if (bias) {
      x0 += bias[(size_t)m * NKQV + n0];
      x1 += bias[(size_t)m * NKQV + n0 + 16];
    }
    out[(size_t)m * NKQV + n0]      = x0;
    out[(size_t)m * NKQV + n0 + 16] = x1;
  }
}

// ---------------------------------------------------------------------------
// Causal distance-attention, single pass (weights bounded by 1, no online max):
//   out[t] = (sum_{s<=t} exp(-max(d2,0)) * softsign(v_s)) / (0.25 + rowsum)
// Each wave owns 16 query rows; k/v 16-key tiles staged by per-wave TDM
// (TENSORcnt is per-wave -> no barriers), double-buffered.
// k^T B-operand reads index k[key][hs] directly, so no transpose staging.
// ---------------------------------------------------------------------------
__global__ __launch_bounds__(256) void attn_wmma_f32(
    const float* __restrict__ pre,  // (B*T) x 192 : [k | q | v]
    float* __restrict__ out) {      // (B*T) x 64
  __shared__ float kts[8][2][16][68];  // per-wave k tile [key][hs], TDM pad 64+4
  __shared__ float vts[8][2][16][68];  // per-wave raw v tile [key][hs]
  __shared__ float wscr[8][16][17];    // per-wave wei tile (C layout -> A layout)
  __shared__ float qqs[8][16];         // per-wave |q|^2 per row

  const int lane = threadIdx.x & 31;
  const int wave = threadIdx.x >> 5;
  const int half = lane >> 4;
  const int lr   = lane & 15;
  const int b    = blockIdx.y;
  const int qbase = blockIdx.x * 128 + wave * 16;  // first query row (t) of this wave
  const size_t rowb = (size_t)b * TLEN;
  const int ntiles = (qbase >> 4) + 1;

  auto issue = [&](int c) {
    const int buf = c & 1;
    const float* base = &pre[(rowb + (size_t)(c << 4)) * NKQV];
    tdm_load_2d(&kts[wave][buf][0][0], base,          64, 16, NKQV, 5, 3, lane);
    tdm_load_2d(&vts[wave][buf][0][0], base + 2 * HS, 64, 16, NKQV, 5, 3, lane);
  };

  issue(0);  // overlap first DMA with q-fragment loads below

  // --- q fragments (A layout, K=64 -> 16 chunks) + row |q|^2 ---
  v2f qa[16];
  float qqp = 0.f;
  #pragma unroll
  for (int j = 0; j < 16; ++j) {
    const int k0 = j * 4 + 2 * half;
    const float* p = &pre[(rowb + qbase + lr) * NKQV + HS + k0];
    const float x = p[0], y = p[1];
    qa[j].x = x; qa[j].y = y;
    qqp += x * x + y * y;
  }
  const float qqrow = qqp + __shfl_xor(qqp, 16, 32);  // combine lane halves
  if (half == 0) qqs[wave][lr] = qqrow;
  float qql[8];
  #pragma unroll
  for (int v = 0; v < 8; ++v) qql[v] = qqs[wave][v + 8 * half];

  v8f o0 = {}, o1 = {}, o2 = {}, o3 = {};
  float den[8];
  #pragma unroll
  for (int v = 0; v < 8; ++v) den[v] = 0.f;

  for (int c = 0; c < ntiles; ++c) {
    if (c + 1 < ntiles) { issue(c + 1); tdm_wait(2); }  // cur pair retired (in-order)
    else                { tdm_wait(0); }
    const int buf = c & 1;

    // |k|^2 per key (lane covers key = lr, its half of hs)
    float kkp = 0.f;
    #pragma unroll 8
    for (int j = 0; j < 32; ++j) {
      const float v = kts[wave][buf][lr][half * 32 + j];
      kkp += v * v;
    }
    const float kkv = kkp + __shfl_xor(kkp, 16, 32);  // |k|^2 for key col = lr

    // scores s = q @ k^T   (16 wmma, K=64); b.x/b.y contiguous -> b64 LDS loads
    v8f s = {};
    #pragma unroll
    for (int j = 0; j < 16; ++j) {
      const int k0 = j * 4 + 2 * half;
      v2f kb; kb.x = kts[wave][buf][lr][k0]; kb.y = kts[wave][buf][lr][k0 + 1];
      s = __builtin_amdgcn_wmma_f32_16x16x4_f32(false, qa[j], false, kb, (short)0, s, false, false);
    }

    // d2 -> wei -> LDS (A-layout staging) + row-sum accumulation
    const bool diag = (c == ntiles - 1);
    #pragma unroll
    for (int v = 0; v < 8; ++v) {
      const int m = v + 8 * half;
      const float d2 = qql[v] + kkv - 2.0f * s[v];
      float wei = __expf(-fmaxf(d2, 0.0f));
      if (diag && lr > m) wei = 0.0f;  // causal mask on diagonal tile
      wscr[wave][m][lr] = wei;
      float rs = wei;
      rs += __shfl_xor(rs, 1, 32);
      rs += __shfl_xor(rs, 2, 32);
      rs += __shfl_xor(rs, 4, 32);
      rs += __shfl_xor(rs, 8, 32);   // sum over 16 key columns in this half
      den[v] += rs;
    }

    // out += wei @ softsign(v)   (K=16 -> 4 chunks, N=64 -> 4 tiles: 16 wmma)
    #pragma unroll
    for (int j = 0; j < 4; ++j) {
      const int k0 = j * 4 + 2 * half;
      v2f a2; a2.x = wscr[wave][lr][k0]; a2.y = wscr[wave][lr][k0 + 1];
      #pragma unroll
      for (int t = 0; t < 4; ++t) {
        const float r0 = vts[wave][buf][k0][t * 16 + lr];
        const float r1 = vts[wave][buf][k0 + 1][t * 16 + lr];
        v2f bb;
        bb.x = r0 * rsqrtf(r0 * r0 + 1.0f);
        bb.y = r1 * rsqrtf(r1 * r1 + 1.0f);
        if (t == 0) o0 = __builtin_amdgcn_wmma_f32_16x16x4_f32(false, a2, false, bb, (short)0, o0, false, false);
        if (t == 1) o1 = __builtin_amdgcn_wmma_f32_16x16x4_f32(false, a2, false, bb, (short)0, o1, false, false);
        if (t == 2) o2 = __builtin_amdgcn_wmma_f32_16x16x4_f32(false, a2, false, bb, (short)0, o2, false, false);
        if (t == 3) o3 = __builtin_amdgcn_wmma_f32_16x16x4_f32(false, a2, false, bb, (short)0, o3, false, false);
      }
    }
  }

  // normalize + store (row m = v + 8*half, cols = {0,16,32,48}+lr)
  #pragma unroll
  for (int v = 0; v < 8; ++v) {
    const int t = qbase + v + 8 * half;
    const float inv = 1.0f / (0.25f + den[v]);
    const size_t ro = (rowb + t) * (size_t)HS;
    out[ro + lr]      = o0[v] * inv;
    out[ro + 16 + lr] = o1[v] * inv;
    out[ro + 32 + lr] = o2[v] * inv;
    out[ro + 48 + lr] = o3[v] * inv;
  }
}

// ---------------------------------------------------------------------------
// Host orchestration
// ---------------------------------------------------------------------------
extern "C" void kernel_launch(void* const* d_in, const int* in_sizes, int n_in,
                              void* d_out, int out_size, void* d_ws, size_t ws_size,
                              hipStream_t stream) {
  (void)in_sizes; (void)n_in; (void)out_size; (void)ws_size;
  const float* x    = (const float*)d_in[0];  // (16,1024,768)
  const float* A    = (const float*)d_in[1];  // (256,256)
  const float* Bmat = (const float*)d_in[2];  // (768,256)
  float* dout = (float*)d_out;                // (16,1024,64)

  float* ws   = (float*)d_ws;
  float* An   = ws;                                   // 256*256
  float* BU   = An + 256 * 256;                       // MROWS*192  (== pre_1, and bias)
  float* preA = BU + (size_t)MROWS * NKQV;            // MROWS*192
  float* preB = preA + (size_t)MROWS * NKQV;          // MROWS*192
  float* outA = preB + (size_t)MROWS * NKQV;          // MROWS*64

  normalize_A<<<dim3(1), dim3(256), 0, stream>>>(A, An);

  const dim3 ggrid(NKQV / 64, MROWS / 64), gblk(256);
  // B_U = x @ Bmat[:, :192]
  gemm_wmma_f32<<<ggrid, gblk, 0, stream>>>(x, 768, 768, nullptr, 0,
                                            Bmat, 256, nullptr, BU, 768);

  const dim3 agrid(TLEN / 128, BATCH), ablk(256);
  // iter 1: X0 = 0  =>  pre_1 = B_U
  attn_wmma_f32<<<agrid, ablk, 0, stream>>>(BU, outA);

  const float* curPre = BU;
  float* curOut = outA;
  float* preBuf[2] = {preA, preB};
  int flip = 0;
  for (int i = 2; i <= 8; ++i) {
    float* nPre = preBuf[flip]; flip ^= 1;
    float* nOut = (i == 8) ? dout : outA;  // stream-ordered: gemm consumed curOut first
    // pre_i = [pre_{i-1} | out_{i-1}] @ An[:, :192] + B_U
    gemm_wmma_f32<<<ggrid, gblk, 0, stream>>>(curPre, NKQV, NKQV, curOut, HS,
                                              An, 256, BU, nPre, 256);
    attn_wmma_f32<<<agrid, ablk, 0, stream>>>(nPre, nOut);
    curPre = nPre; curOut = nOut;
  }
}